// MPNNConv_14173392077056
// MI455X (gfx1250) — compile-verified
//
#include <hip/hip_runtime.h>
#include <math.h>

// GATConv forward for MI455X (gfx1250, wave32).
// Pipeline: [WMMA f32 GEMM] -> [attn dots] -> [init out/max/denom]
//           -> [edge max (atomic)] -> [edge exp + denom (atomic)] -> [aggregate (atomic)]

typedef __attribute__((ext_vector_type(2))) float v2f;
typedef __attribute__((ext_vector_type(8))) float v8f;

#define IN_FEATS 128
#define HEADS 2
#define OUTF 32
#define HD (HEADS * OUTF) /* 64 */
#define NEG_SLOPE 0.2f

// Exact float atomic-max via int-max (nonneg) / uint-min (neg). Order-independent.
__device__ __forceinline__ void atomicMaxF(float* addr, float val) {
    if (val >= 0.0f) atomicMax((int*)addr, __float_as_int(val));
    else             atomicMin((unsigned int*)addr, __float_as_uint(val));
}

__device__ __forceinline__ float leaky(float v) {
    return v > 0.0f ? v : NEG_SLOPE * v;
}

// ---------------------------------------------------------------------------
// Kernel 1: ft = feat @ fc_w^T  via V_WMMA_F32_16X16X4_F32 (exact fp32).
// One wave -> 16 (M) x 64 (N) output tile. K=128 in 32 steps of 4.
// A fragment (16x4): lanes 0-15 hold K={k0,k0+1}, lanes 16-31 hold K={k0+2,k0+3}.
// B fragment (4x16): lane half selects K pair, lane%16 selects N column.
// C/D (16x16): VGPR r -> row r (lanes 0-15) / row r+8 (lanes 16-31), col = lane%16.
// ---------------------------------------------------------------------------
__global__ __launch_bounds__(256) void gat_gemm_wmma(
        const float* __restrict__ feat, const float* __restrict__ fcw,
        float* __restrict__ ft, int n_nodes) {
    const int lane = threadIdx.x & 31;
    const int wave = blockIdx.x * (blockDim.x >> 5) + (threadIdx.x >> 5);
    const int row0 = wave * 16;
    if (row0 >= n_nodes) return;

    const int hsel = lane >> 4;   // lane half: selects K pair within k-step
    const int l16  = lane & 15;

    int arow = row0 + l16;
    if (arow >= n_nodes) arow = n_nodes - 1;   // clamp loads; stores are guarded
    const float* __restrict__ A  = feat + (size_t)arow * IN_FEATS;
    const float* __restrict__ B0 = fcw + (size_t)(l16 +  0) * IN_FEATS;
    const float* __restrict__ B1 = fcw + (size_t)(l16 + 16) * IN_FEATS;
    const float* __restrict__ B2 = fcw + (size_t)(l16 + 32) * IN_FEATS;
    const float* __restrict__ B3 = fcw + (size_t)(l16 + 48) * IN_FEATS;

    v8f c0 = {}, c1 = {}, c2 = {}, c3 = {};
    #pragma unroll 8
    for (int k0 = 0; k0 < IN_FEATS; k0 += 4) {
        const int k = k0 + 2 * hsel;
        v2f a, b0, b1, b2, b3;
        a.x  = A[k];  a.y  = A[k + 1];
        b0.x = B0[k]; b0.y = B0[k + 1];
        b1.x = B1[k]; b1.y = B1[k + 1];
        b2.x = B2[k]; b2.y = B2[k + 1];
        b3.x = B3[k]; b3.y = B3[k + 1];
        c0 = __builtin_amdgcn_wmma_f32_16x16x4_f32(false, a, false, b0, (short)0, c0, false, false);
        c1 = __builtin_amdgcn_wmma_f32_16x16x4_f32(false, a, false, b1, (short)0, c1, false, false);
        c2 = __builtin_amdgcn_wmma_f32_16x16x4_f32(false, a, false, b2, (short)0, c2, false, false);
        c3 = __builtin_amdgcn_wmma_f32_16x16x4_f32(false, a, false, b3, (short)0, c3, false, false);
    }

    #pragma unroll
    for (int r = 0; r < 8; ++r) {
        const int m = row0 + r + 8 * hsel;
        if (m < n_nodes) {
            float* o = ft + (size_t)m * HD + l16;
            o[0]  = c0[r];
            o[16] = c1[r];
            o[32] = c2[r];
            o[48] = c3[r];
        }
    }
}

// ---------------------------------------------------------------------------
// Kernel 2: el/er[node][h] = dot(ft[node][h][:], attn_{l,r}[h][:])
// ---------------------------------------------------------------------------
__global__ __launch_bounds__(256) void gat_attn_lr(
        const float* __restrict__ ft,
        const float* __restrict__ attn_l, const float* __restrict__ attn_r,
        float* __restrict__ el, float* __restrict__ er, int n_nodes) {
    const int t = blockIdx.x * blockDim.x + threadIdx.x;
    if (t >= n_nodes * HEADS) return;
    const int node = t >> 1, h = t & 1;
    const float* __restrict__ f = ft + (size_t)node * HD + h * OUTF;
    const float* __restrict__ al = attn_l + h * OUTF;
    const float* __restrict__ ar = attn_r + h * OUTF;
    float sl = 0.0f, sr = 0.0f;
    #pragma unroll
    for (int d = 0; d < OUTF; ++d) {
        const float v = f[d];
        sl += v * al[d];
        sr += v * ar[d];
    }
    el[t] = sl;
    er[t] = sr;
}

// ---------------------------------------------------------------------------
// Kernel 3: init out <- bias (broadcast), maxv <- -inf, denom <- 0
// ---------------------------------------------------------------------------
__global__ __launch_bounds__(256) void gat_init(
        float* __restrict__ out, const float* __restrict__ bias,
        float* __restrict__ maxv, float* __restrict__ denom, int n_nodes) {
    const int t = blockIdx.x * blockDim.x + threadIdx.x;
    if (t < n_nodes * HD)    out[t] = bias[t & (HD - 1)];
    if (t < n_nodes * HEADS) { maxv[t] = -INFINITY; denom[t] = 0.0f; }
}

// ---------------------------------------------------------------------------
// Kernel 4: per-(edge,head) segment max into maxv[dst][h]
// ---------------------------------------------------------------------------
__global__ __launch_bounds__(256) void gat_edge_max(
        const int* __restrict__ src, const int* __restrict__ dst,
        const float* __restrict__ el, const float* __restrict__ er,
        float* __restrict__ maxv, int n_edges) {
    const int t = blockIdx.x * blockDim.x + threadIdx.x;
    if (t >= n_edges * HEADS) return;
    const int e = t >> 1, h = t & 1;
    const int s = src[e], d = dst[e];
    const float v = leaky(el[s * HEADS + h] + er[d * HEADS + h]);
    atomicMaxF(&maxv[d * HEADS + h], v);
}

// ---------------------------------------------------------------------------
// Kernel 5: alpha[e][h] = exp(e - max); denom[dst][h] += alpha
// ---------------------------------------------------------------------------
__global__ __launch_bounds__(256) void gat_edge_exp(
        const int* __restrict__ src, const int* __restrict__ dst,
        const float* __restrict__ el, const float* __restrict__ er,
        const float* __restrict__ maxv, float* __restrict__ denom,
        float* __restrict__ alpha, int n_edges) {
    const int t = blockIdx.x * blockDim.x + threadIdx.x;
    if (t >= n_edges * HEADS) return;
    const int e = t >> 1, h = t & 1;
    const int s = src[e], d = dst[e];
    const float v = leaky(el[s * HEADS + h] + er[d * HEADS + h]);
    const float ex = expf(v - maxv[d * HEADS + h]);
    alpha[t] = ex;
    atomicAdd(&denom[d * HEADS + h], ex);
}

// ---------------------------------------------------------------------------
// Kernel 6: out[dst][h][:] += ft[src][h][:] * (alpha / denom[dst][h]).
// 64 threads per edge -> coalesced 256B gather + coalesced atomic scatter.
// ---------------------------------------------------------------------------
__global__ __launch_bounds__(256) void gat_aggregate(
        const int* __restrict__ src, const int* __restrict__ dst,
        const float* __restrict__ ft, const float* __restrict__ alpha,
        const float* __restrict__ denom, float* __restrict__ out, int n_edges) {
    const int t = blockIdx.x * blockDim.x + threadIdx.x;
    if (t >= n_edges * HD) return;
    const int e = t >> 6, c = t & (HD - 1), h = c >> 5;
    const int s = src[e], d = dst[e];
    const float a = alpha[e * HEADS + h] / denom[d * HEADS + h];
    atomicAdd(&out[(size_t)d * HD + c], ft[(size_t)s * HD + c] * a);
}

extern "C" void kernel_launch(void* const* d_in, const int* in_sizes, int n_in,
                              void* d_out, int out_size, void* d_ws, size_t ws_size,
                              hipStream_t stream) {
    // setup_inputs order: feat, src, dst, fc_w, attn_l, attn_r, bias
    const float* feat   = (const float*)d_in[0];
    const int*   src    = (const int*)d_in[1];
    const int*   dst    = (const int*)d_in[2];
    const float* fcw    = (const float*)d_in[3];
    const float* attn_l = (const float*)d_in[4];
    const float* attn_r = (const float*)d_in[5];
    const float* bias   = (const float*)d_in[6];
    float* out = (float*)d_out;

    const int n_nodes = in_sizes[0] / IN_FEATS;   // 100000
    const int n_edges = in_sizes[1];              // 1600000

    // Workspace carve (floats): ft[N*64] el[N*2] er[N*2] maxv[N*2] denom[N*2] alpha[E*2]
    float* ft    = (float*)d_ws;
    float* el    = ft    + (size_t)n_nodes * HD;
    float* er    = el    + (size_t)n_nodes * HEADS;
    float* maxv  = er    + (size_t)n_nodes * HEADS;
    float* denom = maxv  + (size_t)n_nodes * HEADS;
    float* alpha = denom + (size_t)n_nodes * HEADS;

    // 1) WMMA GEMM: one wave per 16 rows, 8 waves per block
    const int mtiles = (n_nodes + 15) / 16;
    gat_gemm_wmma<<<(mtiles + 7) / 8, 256, 0, stream>>>(feat, fcw, ft, n_nodes);

    // 2) attention dots
    const int nh = n_nodes * HEADS;
    gat_attn_lr<<<(nh + 255) / 256, 256, 0, stream>>>(ft, attn_l, attn_r, el, er, n_nodes);

    // 3) init out/maxv/denom
    const int init_n = n_nodes * HD;
    gat_init<<<(init_n + 255) / 256, 256, 0, stream>>>(out, bias, maxv, denom, n_nodes);

    // 4) segment max, 5) exp + segment sum
    const int eh = n_edges * HEADS;
    gat_edge_max<<<(eh + 255) / 256, 256, 0, stream>>>(src, dst, el, er, maxv, n_edges);
    gat_edge_exp<<<(eh + 255) / 256, 256, 0, stream>>>(src, dst, el, er, maxv, denom, alpha, n_edges);

    // 6) weighted scatter-add aggregation (dominant, bandwidth/atomic bound)
    const long long total = (long long)n_edges * HD;
    gat_aggregate<<<(int)((total + 255) / 256), 256, 0, stream>>>(src, dst, ft, alpha, denom, out, n_edges);
}